// LQRO_15736760172585
// MI455X (gfx1250) — compile-verified
//
#include <hip/hip_runtime.h>

// ---------------------------------------------------------------------------
// Batched LQR on MI455X (gfx1250).
// One workgroup (8 wave32) per batch element. Q[b,t] (96x96 f32, 36KB)
// streams through a DOUBLE-BUFFERED LDS tile via async global->LDS copies
// (GLOBAL_LOAD_ASYNC_TO_LDS_B128, ASYNCcnt) when the toolchain exposes the
// builtins; V, F=[A|Bm], cx trajectory stay LDS-resident. All GEMMs use
// V_WMMA_F32_16X16X4_F32 (f32-exact). pinv(Quu) via Newton-Schulz on WMMA.
// Kt/kt spill to d_ws for the forward pass.
// ---------------------------------------------------------------------------

typedef __attribute__((ext_vector_type(2))) float v2f;
typedef __attribute__((ext_vector_type(8))) float v8f;
typedef __attribute__((ext_vector_type(4))) int   v4i;

#define NBATCH   128
#define TSTEPS   64
#define NS       64
#define NC       32
#define NSC      96
#define NTHREADS 256
#define NWAVES   8
#define NEWTON_ITERS 14

// Async global->LDS path (guarded: falls back to synchronous float4 copies).
#if defined(__has_builtin)
#if __has_builtin(__builtin_amdgcn_global_load_async_to_lds_b128) && \
    __has_builtin(__builtin_amdgcn_s_wait_asynccnt)
#define ASYNC_LDS 1
#endif
#endif
#ifndef ASYNC_LDS
#define ASYNC_LDS 0
#endif

// ---- LDS layout (float offsets) -------------------------------------------
#define OFF_Q    0        // 2 x 96x96 = 18432  double-buffered Q tile
#define OFF_F    18432    // 64x96  = 6144   F = [A | Bm]
#define OFF_V    24576    // 64x64  = 4096   Riccati value matrix
#define OFF_CX   28672    // 64x64  = 4096   rollout states cx[t][i]
#define OFF_W    32768    // 64x96  = 6144   W = V*F ; reused for T1 = Quu*Kt
#define OFF_X    38912    // 32x32  = 1024   Newton-Schulz X (Quu_inv)
#define OFF_Y    39936    // 32x32  = 1024
#define OFF_Z    40960    // 32x32  = 1024
#define OFF_K    41984    // 32x64  = 2048   Kt
#define OFF_QT   44032    // 96             qt vector
#define OFF_VV   44128    // 64             v vector
#define OFF_KT   44192    // 32             kt vector
#define OFF_R    44224    // 32             r = qu + Quu*kt
#define OFF_XU   44256    // 96             [x; u]
#define OFF_SX   44352    // 64             forward state
#define OFF_SXN  44416    // 64             next state
#define OFF_SU   44480    // 32             control
#define OFF_RED  44512    // 256            cost reduction
#define OFF_CS   44768    // 40             column sums + scale
#define SMEM_FLOATS 44808                   // ~175 KB LDS (of 320 KB/WGP)

#define QTILE   (NSC * NSC)                 // 9216 floats / 36 KB

// ---------------------------------------------------------------------------
// Q-tile streaming helpers: issue() starts the copy of one 36KB tile into an
// LDS buffer; wait() guarantees this wave's portion has landed (a following
// __syncthreads() makes all portions visible workgroup-wide).
// Builtin prototype (from compiler diagnostic): param0 = int4 addrspace(1)*
// (global source), param1 = LDS destination, then offset + cpol immediates.
// ---------------------------------------------------------------------------
__device__ __forceinline__ void q_tile_issue(float* dstLds, const float* src, int tid)
{
#if ASYNC_LDS
    __attribute__((address_space(1))) v4i* g =
        (__attribute__((address_space(1))) v4i*)(void*)const_cast<float*>(src);
    __attribute__((address_space(3))) v4i* l =
        (__attribute__((address_space(3))) v4i*)(void*)dstLds;
    for (int i = tid; i < QTILE / 4; i += NTHREADS) {
        __builtin_amdgcn_global_load_async_to_lds_b128(g + i, l + i, 0, 0);
    }
#else
    const float4* s4 = reinterpret_cast<const float4*>(src);
    float4* d4 = reinterpret_cast<float4*>(dstLds);
    for (int i = tid; i < QTILE / 4; i += NTHREADS) d4[i] = s4[i];
#endif
}

__device__ __forceinline__ void q_tile_wait()
{
#if ASYNC_LDS
    __builtin_amdgcn_s_wait_asynccnt(0);
#endif
}

// ---------------------------------------------------------------------------
// Generic LDS-resident GEMM: C(MxN) = alpha * op(A)(MxK) * B(KxN) [+ C]
// op(A) = A or A^T (transA indexes the physical array transposed; lda is the
// PHYSICAL leading dim). One 16x16 tile per wave, K consumed 4 at a time with
// two interleaved WMMA accumulators. Requires M,N multiples of 16, K of 8.
// Wave-uniform control flow => EXEC all-ones at every WMMA (ISA requirement).
// ---------------------------------------------------------------------------
__device__ __forceinline__ void gemm_lds(const float* __restrict__ A, int lda, bool transA,
                                         const float* __restrict__ B, int ldb,
                                         float* __restrict__ C, int ldc,
                                         int M, int N, int K,
                                         float alpha, bool accum,
                                         int wave, int lane)
{
    const int mt = M >> 4, nt = N >> 4;
    const int half = lane >> 4;          // 0: lanes 0-15, 1: lanes 16-31
    const int m_in = lane & 15;          // A row within tile
    const int n_in = lane & 15;          // B/C col within tile
    const int kb   = half << 1;          // K pair base inside a 4-chunk
    const int mc   = half << 3;          // C row base (0 or 8) per ISA layout

    for (int t = wave; t < mt * nt; t += NWAVES) {
        const int tm = (t / nt) << 4;
        const int tn = (t % nt) << 4;
        const int r  = tm + m_in;
        const int cn = tn + n_in;
        v8f acc0 = {0.f, 0.f, 0.f, 0.f, 0.f, 0.f, 0.f, 0.f};
        v8f acc1 = {0.f, 0.f, 0.f, 0.f, 0.f, 0.f, 0.f, 0.f};
        for (int k0 = 0; k0 < K; k0 += 8) {
            {
                const int ka = k0 + kb;
                v2f a, b;
                a.x = transA ? A[ka * lda + r]       : A[r * lda + ka];
                a.y = transA ? A[(ka + 1) * lda + r] : A[r * lda + ka + 1];
                b.x = B[ka * ldb + cn];
                b.y = B[(ka + 1) * ldb + cn];
                acc0 = __builtin_amdgcn_wmma_f32_16x16x4_f32(
                    false, a, false, b, (short)0, acc0, false, false);
            }
            {
                const int ka = k0 + 4 + kb;
                v2f a, b;
                a.x = transA ? A[ka * lda + r]       : A[r * lda + ka];
                a.y = transA ? A[(ka + 1) * lda + r] : A[r * lda + ka + 1];
                b.x = B[ka * ldb + cn];
                b.y = B[(ka + 1) * ldb + cn];
                acc1 = __builtin_amdgcn_wmma_f32_16x16x4_f32(
                    false, a, false, b, (short)0, acc1, false, false);
            }
        }
#pragma unroll
        for (int v = 0; v < 8; ++v) {
            const float val = alpha * (acc0[v] + acc1[v]);
            float* cp = &C[(tm + mc + v) * ldc + cn];
            *cp = accum ? (*cp + val) : val;
        }
    }
}

__global__ __launch_bounds__(NTHREADS, 1)
void lqr_kernel(const float* __restrict__ x_init,
                const float* __restrict__ Qg,
                const float* __restrict__ pg,
                const float* __restrict__ Ag,
                const float* __restrict__ Bg,
                float* __restrict__ out_x,
                float* __restrict__ out_u,
                float* __restrict__ out_cost,
                float* __restrict__ wsK,
                float* __restrict__ wsk)
{
    extern __shared__ float sm[];
    float* smQ = sm + OFF_Q;      // two QTILE buffers
    float* sF  = sm + OFF_F;
    float* sV  = sm + OFF_V;
    float* scx = sm + OFF_CX;
    float* sW  = sm + OFF_W;
    float* sT1 = sm + OFF_W;      // alias: W is dead once Qt is formed
    float* sX  = sm + OFF_X;
    float* sY  = sm + OFF_Y;
    float* sZ  = sm + OFF_Z;
    float* sK  = sm + OFF_K;
    float* sqt = sm + OFF_QT;
    float* svv = sm + OFF_VV;
    float* skt = sm + OFF_KT;
    float* sr  = sm + OFF_R;
    float* sxu = sm + OFF_XU;
    float* sx  = sm + OFF_SX;
    float* sxn = sm + OFF_SXN;
    float* su  = sm + OFF_SU;
    float* sred= sm + OFF_RED;
    float* scs = sm + OFF_CS;

    const int tid  = threadIdx.x;
    const int lane = tid & 31;
    const int wave = tid >> 5;
    const int b    = blockIdx.x;
    const float* Qbase = Qg + (size_t)b * TSTEPS * QTILE;

    // ---- Phase 0: load F = [A | Bm], zero V & v, uncontrolled rollout -----
    for (int e = tid; e < NS * NSC; e += NTHREADS) {
        const int i = e / NSC, j = e % NSC;
        sF[e] = (j < NS) ? Ag[i * NS + j] : Bg[i * NC + (j - NS)];
    }
    for (int e = tid; e < NS * NS; e += NTHREADS) sV[e] = 0.f;
    if (tid < NS) {
        svv[tid] = 0.f;
        scx[tid] = x_init[(size_t)b * NS + tid];
    }
    __syncthreads();
    for (int t = 1; t < TSTEPS; ++t) {               // cx[t] = cx[t-1] @ A^T
        if (tid < NS) {
            float a = 0.f;
            const float* xp = &scx[(t - 1) * NS];
            const float* ar = &sF[tid * NSC];        // A row (first 64 cols of F)
            for (int j = 0; j < NS; ++j) a += xp[j] * ar[j];
            scx[t * NS + tid] = a;
        }
        __syncthreads();
    }

    // ---- Phase 1: backward Riccati sweep ---------------------------------
    // Double-buffered: tile t-1 streams into LDS while tile t is crunched.
    q_tile_issue(&smQ[((TSTEPS - 1) & 1) * QTILE], Qbase + (size_t)(TSTEPS - 1) * QTILE, tid);
    for (int t = TSTEPS - 1; t >= 0; --t) {
        float* sQ = &smQ[(t & 1) * QTILE];
        q_tile_wait();
        __syncthreads();                              // tile t visible to all waves
        if (t > 0)                                    // overlap next tile with compute
            q_tile_issue(&smQ[((t - 1) & 1) * QTILE], Qbase + (size_t)(t - 1) * QTILE, tid);

        // qt = p + Q*[cx;0] + F^T v   (fused p_new)
        if (tid < NSC) {
            const float* qrow = &sQ[tid * NSC];
            const float* cxt  = &scx[t * NS];
            float a = pg[((size_t)b * TSTEPS + t) * NSC + tid];
            for (int j = 0; j < NS; ++j) a += qrow[j] * cxt[j];
            float a2 = 0.f;
            for (int k = 0; k < NS; ++k) a2 += sF[k * NSC + tid] * svv[k];
            sqt[tid] = a + a2;
        }
        __syncthreads();

        // W = V * F   (64x96 = 64x64 * 64x96)
        gemm_lds(sV, NS, false, sF, NSC, sW, NSC, NS, NSC, NS, 1.f, false, wave, lane);
        __syncthreads();
        // Qt = Q + F^T * W  (96x96 += 96x64 * 64x96), in place over sQ
        gemm_lds(sF, NSC, true, sW, NSC, sQ, NSC, NSC, NSC, NS, 1.f, true, wave, lane);
        __syncthreads();

        // ---- pinv(Quu) via Newton-Schulz; Quu = sQ[64:,64:] (SPD) --------
        if (tid < NC) {
            float s = 0.f;
            for (int i = 0; i < NC; ++i) s += fabsf(sQ[(NS + i) * NSC + NS + tid]);
            scs[tid] = s;
        }
        __syncthreads();
        if (tid == 0) {
            float m = 0.f;
            for (int j = 0; j < NC; ++j) m = fmaxf(m, scs[j]);
            scs[32] = 1.0f / (m * m);      // 1/(||Quu||_1 * ||Quu||_inf)
        }
        __syncthreads();
        {
            const float sc = scs[32];
            for (int e = tid; e < NC * NC; e += NTHREADS) {
                const int i = e >> 5, j = e & 31;
                sX[e] = sc * sQ[(NS + i) * NSC + NS + j];   // X0 (symmetric)
            }
        }
        __syncthreads();
        for (int it = 0; it < NEWTON_ITERS; ++it) {
            gemm_lds(&sQ[NS * NSC + NS], NSC, false, sX, NC, sY, NC,
                     NC, NC, NC, 1.f, false, wave, lane);           // Y = Quu*X
            __syncthreads();
            gemm_lds(sX, NC, false, sY, NC, sZ, NC,
                     NC, NC, NC, 1.f, false, wave, lane);           // Z = X*Y
            __syncthreads();
            for (int e = tid; e < NC * NC; e += NTHREADS)
                sX[e] = 2.f * sX[e] - sZ[e];                        // X = 2X - Z
            __syncthreads();
        }

        // Kt = -Quu_inv * Qux   (32x64)
        gemm_lds(sX, NC, false, &sQ[NS * NSC], NSC, sK, NS,
                 NC, NS, NC, -1.f, false, wave, lane);
        // kt = -Quu_inv * qu
        if (tid < NC) {
            float a = 0.f;
            for (int j = 0; j < NC; ++j) a += sX[tid * NC + j] * sqt[NS + j];
            skt[tid] = -a;
        }
        __syncthreads();

        // ---- Vn = Qxx + Qxu*Kt + Kt^T*Qux + Kt^T*(Quu*Kt) ----------------
        for (int e = tid; e < NS * NS; e += NTHREADS) {
            const int i = e >> 6, j = e & 63;
            sV[e] = sQ[i * NSC + j];                                // Qxx
        }
        __syncthreads();
        gemm_lds(&sQ[NS], NSC, false, sK, NS, sV, NS,
                 NS, NS, NC, 1.f, true, wave, lane);                // += Qxu*Kt
        __syncthreads();
        gemm_lds(sK, NS, true, &sQ[NS * NSC], NSC, sV, NS,
                 NS, NS, NC, 1.f, true, wave, lane);                // += Kt^T*Qux
        __syncthreads();
        gemm_lds(&sQ[NS * NSC + NS], NSC, false, sK, NS, sT1, NS,
                 NC, NS, NC, 1.f, false, wave, lane);               // T1 = Quu*Kt
        __syncthreads();
        gemm_lds(sK, NS, true, sT1, NS, sV, NS,
                 NS, NS, NC, 1.f, true, wave, lane);                // += Kt^T*T1
        __syncthreads();

        // vn = qx + Qxu*kt + Kt^T*(qu + Quu*kt)
        if (tid < NC) {
            float a = sqt[NS + tid];
            for (int j = 0; j < NC; ++j) a += sQ[(NS + tid) * NSC + NS + j] * skt[j];
            sr[tid] = a;
        }
        __syncthreads();
        if (tid < NS) {
            float a = sqt[tid];
            for (int j = 0; j < NC; ++j) a += sQ[tid * NSC + NS + j] * skt[j];
            for (int c = 0; c < NC; ++c) a += sK[c * NS + tid] * sr[c];
            svv[tid] = a;
        }

        // spill Kt / kt for forward pass
        {
            float4* d = reinterpret_cast<float4*>(&wsK[((size_t)b * TSTEPS + t) * (NC * NS)]);
            const float4* s = reinterpret_cast<const float4*>(sK);
            for (int i = tid; i < (NC * NS) / 4; i += NTHREADS) d[i] = s[i];
            if (tid < NC) wsk[((size_t)b * TSTEPS + t) * NC + tid] = skt[tid];
        }
    }

    // ---- Phase 2: forward pass (second and last pass over Q) --------------
    if (tid < NS) sx[tid] = x_init[(size_t)b * NS + tid];
    float costPart = 0.f;
    __syncthreads();
    q_tile_issue(&smQ[0], Qbase, tid);
    for (int t = 0; t < TSTEPS; ++t) {
        float* sQ = &smQ[(t & 1) * QTILE];
        q_tile_wait();
        __syncthreads();                              // Q[t] landed; prior users done
        if (t + 1 < TSTEPS)
            q_tile_issue(&smQ[((t + 1) & 1) * QTILE], Qbase + (size_t)(t + 1) * QTILE, tid);
        {
            const float4* k4 = reinterpret_cast<const float4*>(
                &wsK[((size_t)b * TSTEPS + t) * (NC * NS)]);
            float4* dk = reinterpret_cast<float4*>(sK);
            for (int i = tid; i < (NC * NS) / 4; i += NTHREADS) dk[i] = k4[i];
            if (tid < NC) skt[tid] = wsk[((size_t)b * TSTEPS + t) * NC + tid];
        }
        if (t + 1 < TSTEPS) {   // warm L2/WGP$ for the next gain tile
            const char* nk = (const char*)&wsK[((size_t)b * TSTEPS + t + 1) * (NC * NS)];
            __builtin_prefetch(nk + tid * 32, 0, 1);
        }
        __syncthreads();
        if (tid < NC) {                                  // u = Kt*(x-cx) + kt
            float a = skt[tid];
            const float* kr  = &sK[tid * NS];
            const float* cxt = &scx[t * NS];
            for (int j = 0; j < NS; ++j) a += kr[j] * (sx[j] - cxt[j]);
            su[tid] = a;
        }
        __syncthreads();
        if (tid < NS) {
            sxu[tid] = sx[tid];
            out_x[((size_t)b * TSTEPS + t) * NS + tid] = sx[tid];
        } else if (tid < NSC) {
            sxu[tid] = su[tid - NS];
            out_u[((size_t)b * TSTEPS + t) * NC + (tid - NS)] = su[tid - NS];
        }
        __syncthreads();
        if (tid < NSC) {                 // cost += 0.5*xu^T Q xu + xu.p
            const float* qrow = &sQ[tid * NSC];
            float y = 0.f;
            for (int j = 0; j < NSC; ++j) y += qrow[j] * sxu[j];
            costPart += sxu[tid] * (0.5f * y + pg[((size_t)b * TSTEPS + t) * NSC + tid]);
        }
        if (tid < NS) {                  // xn = F * xu = x A^T + u Bm^T
            const float* fr = &sF[tid * NSC];
            float a = 0.f;
            for (int j = 0; j < NSC; ++j) a += fr[j] * sxu[j];
            sxn[tid] = a;
        }
        __syncthreads();
        if (tid < NS) sx[tid] = sxn[tid];
    }

    sred[tid] = costPart;
    __syncthreads();
    for (int s = NTHREADS / 2; s > 0; s >>= 1) {
        if (tid < s) sred[tid] += sred[tid + s];
        __syncthreads();
    }
    if (tid == 0) out_cost[b] = sred[0];
}

extern "C" void kernel_launch(void* const* d_in, const int* in_sizes, int n_in,
                              void* d_out, int out_size, void* d_ws, size_t ws_size,
                              hipStream_t stream) {
    const float* x_init = (const float*)d_in[0];
    const float* Q      = (const float*)d_in[1];
    const float* p      = (const float*)d_in[2];
    const float* A      = (const float*)d_in[3];
    const float* Bm     = (const float*)d_in[4];
    // d_in[5] = dt (unused; reference ignores it)

    float* out      = (float*)d_out;
    float* out_x    = out;                                   // (NB,T,NS)
    float* out_u    = out_x + (size_t)NBATCH * TSTEPS * NS;  // (NB,T,NC)
    float* out_cost = out_u + (size_t)NBATCH * TSTEPS * NC;  // (NB,)

    float* wsK = (float*)d_ws;                               // (NB,T,NC,NS)
    float* wsk = wsK + (size_t)NBATCH * TSTEPS * NC * NS;    // (NB,T,NC)

    const size_t smem = (size_t)SMEM_FLOATS * sizeof(float); // ~175 KB LDS
    lqr_kernel<<<dim3(NBATCH), dim3(NTHREADS), smem, stream>>>(
        x_init, Q, p, A, Bm, out_x, out_u, out_cost, wsK, wsk);
}